// CTRGC_39109972197892
// MI455X (gfx1250) — compile-verified
//
#include <hip/hip_runtime.h>

// Problem constants (match reference)
#define NB 256
#define C  64
#define T  128
#define V  25
#define R  8
#define O  64

#define TC 16              // T-chunk per block
#define M1 (TC*V)          // 400: flattened (t,v) positions per chunk (GEMM1 N-dim)
#define X3_VS 28           // v stride in x3 LDS tile, padded 25->28 (zero pad = K padding)
#define X3_TS (TC*X3_VS)   // 448 floats per o_local

// LDS plan (floats)
#define OFF_XM 0           // 1600  : xm[n] = mean_T(x)  [C][V]
#define OFF_X1 1600        // 200   : x1[r][u]
#define OFF_X2 1800        // 200   : x2[r][v]
#define OFF_D  2000        // 5000  : d[r][u][v] = tanh(x1-x2)
#define OFF_XS 7040        // 25600 : x tile [C][M1]  (filled by TDM)
#define OFF_X3 32640       // 7168  : x3 tile [16][X3_TS] (current o-chunk)
#define SMEM_FLOATS 39808  // ~155.5 KB

typedef float v2f __attribute__((ext_vector_type(2)));
typedef float v8f __attribute__((ext_vector_type(8)));
typedef unsigned int v4u __attribute__((ext_vector_type(4)));
typedef int v4i __attribute__((ext_vector_type(4)));
typedef int v8i __attribute__((ext_vector_type(8)));

__device__ __forceinline__ v8f wmma_f32(v2f a, v2f b, v8f c) {
  // D = A(16x4,f32) x B(4x16,f32) + C(16x16,f32)
  return __builtin_amdgcn_wmma_f32_16x16x4_f32(false, a, false, b, (short)0, c,
                                               false, false);
}

// ---------------------------------------------------------------------------
// Kernel 1: xm[n,c,v] = mean over T of x[n,c,t,v]
// float4 (b128) coalesced reads; LDS float atomics (ds_add_f32) do the
// transpose-sum. A 4-elem group never crosses a c-row (T*V % 4 == 0).
// ---------------------------------------------------------------------------
__global__ void __launch_bounds__(256) ctrgc_mean(const float* __restrict__ x,
                                                  float* __restrict__ xm) {
  __shared__ float xml[C * V];
  const int n = blockIdx.x;
  const int tid = threadIdx.x;
  for (int i = tid; i < C * V; i += 256) xml[i] = 0.f;
  __syncthreads();
  const float4* xn4 =
      reinterpret_cast<const float4*>(x + (size_t)n * C * T * V);
  for (int q = tid; q < (C * T * V) / 4; q += 256) {
    float4 val = xn4[q];
    int idx = q * 4;
    int c = idx / (T * V);
    int v = idx % V;  // T*V is a multiple of V, so this is the v coordinate
    float* row = &xml[c * V];
    atomicAdd(&row[v], val.x); if (++v == V) v = 0;
    atomicAdd(&row[v], val.y); if (++v == V) v = 0;
    atomicAdd(&row[v], val.z); if (++v == V) v = 0;
    atomicAdd(&row[v], val.w);
  }
  __syncthreads();
  const float invT = 1.0f / (float)T;
  for (int i = tid; i < C * V; i += 256)
    xm[(size_t)n * C * V + i] = xml[i] * invT;
}

// ---------------------------------------------------------------------------
// Kernel 2: fused conv3 (GEMM over C) + dynamic-adjacency aggregation (GEMM
// over V). The x-tile (64 x 400 f32, row stride 3200 f32) is staged into LDS
// by the Tensor Data Mover while the block computes x1/x2/d; adjacency is
// built on the fly from d in LDS. All WMMA in fp32 (16x16x4).
// ---------------------------------------------------------------------------
__global__ void __launch_bounds__(512) ctrgc_main(
    const float* __restrict__ x,  const float* __restrict__ Aadj,
    const float* __restrict__ w1, const float* __restrict__ b1,
    const float* __restrict__ w2, const float* __restrict__ b2,
    const float* __restrict__ w3, const float* __restrict__ b3,
    const float* __restrict__ w4, const float* __restrict__ b4,
    const float* __restrict__ xm, float* __restrict__ out) {
  extern __shared__ float sm[];
  float* xms = sm + OFF_XM;
  float* x1s = sm + OFF_X1;
  float* x2s = sm + OFF_X2;
  float* dls = sm + OFF_D;
  float* xs  = sm + OFF_XS;
  float* x3s = sm + OFF_X3;

  const int tcid = blockIdx.x;        // 0 .. T/TC-1
  const int n    = blockIdx.y;        // 0 .. NB-1
  const int t0   = tcid * TC;
  const int tid  = threadIdx.x;
  const int wave = tid >> 5;          // 0..15 (wave32)
  const int lane = tid & 31;
  const int half = lane >> 4;         // 0: lanes 0-15, 1: lanes 16-31
  const int j    = lane & 15;

  // ---- Phase 0a: kick off the TDM load of the x tile (wave 0 issues) ----
  // Tile: 64 rows (c) x 400 f32 (t_local,v), global row stride T*V = 3200 f32.
  if (wave == 0) {
    const float* gsrc = x + ((size_t)n * C * T + t0) * V;
    unsigned long long ga = (unsigned long long)(uintptr_t)gsrc;
    unsigned lds_addr = (unsigned)(uintptr_t)(void*)xs;  // low 32b of flat LDS
    // D# group 0: count=1 | lds_addr | global_addr[56:0] | type=2
    v4u g0 = {1u, lds_addr, (unsigned)ga,
              (unsigned)((ga >> 32) & 0x01FFFFFFu) | (2u << 30)};
    // D# group 1: data_size=4B(code 2); tensor_dim0=400; tensor_dim1=64;
    //             tile_dim0=400; tile_dim1=64; tensor_dim0_stride=3200
    v8i g1 = {(int)(2u << 16),      // [17:16] data_size
              (int)(400u << 16),    // [79:48] tensor_dim0 (lo16 here)
              (int)(64u << 16),     // [95:80] tensor_dim1 lo16 (dim0 hi16 = 0)
              (int)(400u << 16),    // [127:112] tile_dim0 (dim1 hi16 = 0)
              (int)64,              // [143:128] tile_dim1 ; tile_dim2 = 0
              (int)3200,            // [207:160] tensor_dim0_stride lo32
              0, 0};                // stride hi, tensor_dim1_stride = 0 (2D)
    v4i gz4 = {0, 0, 0, 0};         // groups 2/3 unused (<=2D tensor)
    v8i gz8 = {0, 0, 0, 0, 0, 0, 0, 0};
    __builtin_amdgcn_tensor_load_to_lds(g0, g1, gz4, gz4, gz8, 0);
  }

  // ---- Phase 0b: stage xm; zero x3 K-pad; compute x1/x2 and d ----
  for (int i = tid; i < C * V; i += 512)
    xms[i] = xm[(size_t)n * C * V + i];
  for (int i = tid; i < 16 * X3_TS; i += 512) x3s[i] = 0.f;
  __syncthreads();

  // x1[r,u] and x2[r,v] (tiny GEMVs over C)
  if (tid < 2 * R * V) {
    int which = tid / (R * V);
    int idx = tid - which * (R * V);
    int r = idx / V, u = idx - r * V;
    const float* w = which ? w2 : w1;
    float s = which ? b2[r] : b1[r];
    for (int c = 0; c < C; ++c) s += w[r * C + c] * xms[c * V + u];
    (which ? x2s : x1s)[idx] = s;
  }
  __syncthreads();

  for (int i = tid; i < R * V * V; i += 512) {
    int r = i / (V * V);
    int rem = i - r * V * V;
    int u = rem / V, vv = rem - u * V;
    dls[i] = tanhf(x1s[r * V + u] - x2s[r * V + vv]);
  }

  // TDM completion: wave 0 waits on TENSORcnt, then the barrier publishes xs.
  if (wave == 0) __builtin_amdgcn_s_wait_tensorcnt(0);
  __syncthreads();

  // ---- Main loop over o-chunks of 16 output channels ----
  for (int oc = 0; oc < O / 16; ++oc) {
    const int obase = oc * 16;

    // GEMM1: x3[o_l][t][v] = b3[o] + sum_c w3[o,c] * x[c][t,v]
    // A = w3 tile (16x64), B = xs (64 x 400), D tiles of 16x16, K-steps of 4.
    for (int mt = wave; mt < M1 / 16; mt += 16) {
      const int m0 = mt * 16;
      v8f acc = {};
#pragma unroll
      for (int k0 = 0; k0 < C; k0 += 4) {
        const int ka = k0 + 2 * half;       // lane-half K split per ISA layout
        v2f a, b;
        a.x = w3[(obase + j) * C + ka];     // A[M=j][K=ka]
        a.y = w3[(obase + j) * C + ka + 1];
        b.x = xs[ka * M1 + m0 + j];         // B[K=ka][N=j]
        b.y = xs[(ka + 1) * M1 + m0 + j];
        acc = wmma_f32(a, b, acc);
      }
      const int m = m0 + j;                 // D column -> (t,v)
      const int t = m / V, v = m - t * V;
#pragma unroll
      for (int r = 0; r < 8; ++r) {
        const int o_l = r + 8 * half;       // D row
        x3s[o_l * X3_TS + t * X3_VS + v] = acc[r] + b3[obase + o_l];
      }
    }
    __syncthreads();

    // GEMM2: out[t,u] = sum_v adj[o,u,v] * x3[o_l][t,v]
    // A = x3 tile (16 t x 28 v, zero-padded K), B[v][u] built on the fly:
    //   adj(u,v) = b4[o] + A[u,v] + sum_r w4[o,r]*d[r,u,v]
    for (int task = wave; task < 32; task += 16) {
      const int o_l = task >> 1;
      const int ut  = task & 1;
      const int o   = obase + o_l;
      const int u   = ut * 16 + j;
      float w4o[R];
#pragma unroll
      for (int r = 0; r < R; ++r) w4o[r] = w4[o * R + r];
      const float bb = b4[o];
      const int uc = (u < V) ? u : (V - 1);   // clamp for safe loads
      v8f acc = {};
#pragma unroll
      for (int k0 = 0; k0 < X3_VS; k0 += 4) {
        const int ka = k0 + 2 * half;
        v2f a, b;
        a.x = x3s[o_l * X3_TS + j * X3_VS + ka];      // A[M=t=j][K=v=ka]
        a.y = x3s[o_l * X3_TS + j * X3_VS + ka + 1];
        const int k0c = (ka < V) ? ka : (V - 1);
        const int k1c = (ka + 1 < V) ? (ka + 1) : (V - 1);
        float s0 = bb + Aadj[uc * V + k0c];
        float s1 = bb + Aadj[uc * V + k1c];
#pragma unroll
        for (int r = 0; r < R; ++r) {
          s0 += w4o[r] * dls[r * V * V + uc * V + k0c];
          s1 += w4o[r] * dls[r * V * V + uc * V + k1c];
        }
        b.x = (u < V && ka     < V) ? s0 : 0.f;       // selects keep EXEC full
        b.y = (u < V && ka + 1 < V) ? s1 : 0.f;
        acc = wmma_f32(a, b, acc);
      }
#pragma unroll
      for (int r = 0; r < 8; ++r) {
        const int t = r + 8 * half;
        if (u < V)
          out[(((size_t)n * O + o) * T + t0 + t) * V + u] = acc[r];
      }
    }
    __syncthreads();
  }
}

// ---------------------------------------------------------------------------
extern "C" void kernel_launch(void* const* d_in, const int* in_sizes, int n_in,
                              void* d_out, int out_size, void* d_ws,
                              size_t ws_size, hipStream_t stream) {
  const float* x    = (const float*)d_in[0];
  const float* Aadj = (const float*)d_in[1];
  const float* w1   = (const float*)d_in[2];
  const float* b1   = (const float*)d_in[3];
  const float* w2   = (const float*)d_in[4];
  const float* b2   = (const float*)d_in[5];
  const float* w3   = (const float*)d_in[6];
  const float* b3   = (const float*)d_in[7];
  const float* w4   = (const float*)d_in[8];
  const float* b4   = (const float*)d_in[9];
  float* out = (float*)d_out;
  float* xm  = (float*)d_ws;  // N*C*V floats = 1.6 MB scratch

  ctrgc_mean<<<NB, 256, 0, stream>>>(x, xm);

  dim3 grid(T / TC, NB);
  size_t smem = (size_t)SMEM_FLOATS * sizeof(float);  // ~155.5 KB / WGP's 320 KB
  ctrgc_main<<<grid, 512, smem, stream>>>(x, Aadj, w1, b1, w2, b2, w3, b3, w4,
                                          b4, xm, out);
}